// MultiheadSelfAttentionVarSizedElementReduce_50302656971016
// MI455X (gfx1250) — compile-verified
//
#include <hip/hip_runtime.h>

// ---------------------------------------------------------------------------
// Segmented multi-head attention reduce for MI455X (gfx1250, wave32).
//
// Algebraic collapse: per-segment shared query lets us fold Wk into a per-
// sample [256 x 8] "kq" matrix, and fold Wv through the probs-weighted pooled
// matrix P[s*8+h][c].  FLOPs drop ~10x and the kernel is HBM-bound
// (~0.5 GB E reads).  Small dense GEMMs run on V_WMMA_F32_16X16X4_F32 to
// keep full fp32 accuracy.  All WMMA fragment loads are unconditional b64s
// (pad rows zero-filled or address-clamped) so no EXEC churn in the K-loops.
// ---------------------------------------------------------------------------

#define SPB   8        // samples per block (segments are contiguous: map is sorted)
#define MAXM  1024     // max elements per 8-sample group (mean ~244, sigma ~16)
#define TSTR  260      // padded fp32 row stride for LDS tiles (even -> 8B align)
#define KQSTR 72       // kq row stride: 64 cols (8 samples x 8 heads) + pad

typedef __attribute__((ext_vector_type(2))) float v2f;
typedef __attribute__((ext_vector_type(8))) float v8f;

// LDS region offsets (floats), lifetime-overlaid:
//  R1: pooled(16x260)      -> per_sample(16x260)
//  R2: q(16x260)           -> E staging tile(32x260)
//  R3: kq(256x72)          -> P(64x260)
//  R4: scores/probs (MAXM x 8)
#define OFF_R1 0
#define OFF_R2 (OFF_R1 + 16 * TSTR)
#define OFF_R3 (OFF_R2 + 32 * TSTR)
#define OFF_R4 (OFF_R3 + 256 * KQSTR)
#define SMEM_FLOATS (OFF_R4 + MAXM * 8)

__device__ __forceinline__ v8f wmma_f32_k4(v2f a, v2f b, v8f c) {
    // D = A(16x4 f32) * B(4x16 f32) + C(16x16 f32)
    return __builtin_amdgcn_wmma_f32_16x16x4_f32(
        false, a, false, b, (short)0, c, false, false);
}

__device__ __forceinline__ v2f ld2(const float* p) {
    return *(const v2f*)p;   // 8B-aligned by construction
}

__launch_bounds__(256, 2)
__global__ void msa_segment_attn_kernel(const float* __restrict__ E,
                                        const int*   __restrict__ seg,
                                        const float* __restrict__ Wq,
                                        const float* __restrict__ Wk,
                                        const float* __restrict__ Wv,
                                        const float* __restrict__ Wo,
                                        float*       __restrict__ out,
                                        int N, int S) {
    __shared__ __align__(16) float sm[SMEM_FLOATS];
    __shared__ int sstart[SPB + 1];

    const int t    = threadIdx.x;
    const int lane = t & 31;
    const int w    = t >> 5;
    const int s0   = blockIdx.x * SPB;

    // ---- phase 0: binary-search segment boundaries (seg is sorted) --------
    if (t <= SPB) {
        int key = s0 + t; if (key > S) key = S;
        int lo = 0, hi = N;
        while (lo < hi) { int mid = (lo + hi) >> 1; if (seg[mid] < key) lo = mid + 1; else hi = mid; }
        sstart[t] = lo;
    }
    __syncthreads();
    const int e0 = sstart[0], e1 = sstart[SPB];

    float* pool = sm + OFF_R1;   // later: per_sample
    float* qmat = sm + OFF_R2;   // later: E staging tile
    float* tile = sm + OFF_R2;
    float* kq   = sm + OFF_R3;   // later: P
    float* Pm   = sm + OFF_R3;
    float* sc   = sm + OFF_R4;   // scores, overwritten in-place by probs

    // ---- phase 1: segment-mean pooling; thread t owns column c = t --------
    {
        const int c = t;
        for (int i = 0; i < SPB; ++i) {
            const int rs = sstart[i], re = sstart[i + 1];
            float acc = 0.f;
            for (int r = rs; r < re; ++r) acc += E[(size_t)r * 256 + c];
            pool[i * TSTR + c] = acc / fmaxf((float)(re - rs), 1.f);
        }
        // zero pad rows 8..15 so WMMA A-fragments need no predication
        for (int i = SPB; i < 16; ++i) pool[i * TSTR + c] = 0.f;
    }
    __syncthreads();

    const int m   = lane & 15;           // A-row / B-col index for this lane
    const int klo = (lane < 16) ? 0 : 2; // K sub-offset per documented layout

    // ---- phase 2: q(16x256) = pooled @ Wq via WMMA f32 --------------------
    for (int nt = w; nt < 16; nt += 8) {
        const int n = nt * 16 + m;
        v8f acc = {0.f, 0.f, 0.f, 0.f, 0.f, 0.f, 0.f, 0.f};
        const float* arow = pool + m * TSTR + klo;     // pad rows are zero
        for (int kb = 0; kb < 256; kb += 4) {
            v2f a = ld2(arow + kb);                    // ds_load_b64
            v2f b;
            b.x = Wq[(size_t)(kb + klo) * 256 + n];
            b.y = Wq[(size_t)(kb + klo + 1) * 256 + n];
            acc = wmma_f32_k4(a, b, acc);
        }
#pragma unroll
        for (int j = 0; j < 8; ++j) {
            const int row = (lane < 16) ? j : (j + 8); // rows 8..15 get zeros
            qmat[row * TSTR + n] = acc[j];
        }
    }
    __syncthreads();

    // ---- phase 3: kq[c][s*8+h] = (1/sqrt(32)) * Wk_head @ q_head^T --------
    {
        const float SCALE = 0.17677669529663689f; // 1/sqrt(head_dim=32)
        for (int job = w; job < 128; job += 8) {  // 8 heads x 16 c-tiles
            const int h = job >> 4, mt = job & 15;
            const int c = mt * 16 + m;
            const float* arow = Wk + (size_t)c * 256 + h * 32 + klo;
            const float* brow = qmat + m * TSTR + h * 32 + klo; // rows>=SPB are zero
            v8f acc = {0.f, 0.f, 0.f, 0.f, 0.f, 0.f, 0.f, 0.f};
            for (int kb = 0; kb < 32; kb += 4) {
                v2f a = ld2(arow + kb);               // global_load_b64
                v2f b = ld2(brow + kb);               // ds_load_b64
                b.x *= SCALE; b.y *= SCALE;
                acc = wmma_f32_k4(a, b, acc);
            }
#pragma unroll
            for (int j = 0; j < 8; ++j) {
                const int crow = mt * 16 + ((lane < 16) ? j : (j + 8));
                const int s    = m; // D column = sample index
                if (s < SPB) kq[crow * KQSTR + s * 8 + h] = acc[j];
            }
        }
    }
    __syncthreads();

    // ---- phase 4: scores[n][h] = E[n,:] . kq[:, sid,h]  (fp32 scalar) -----
    {
        const int nl = t >> 3, h = t & 7;
        for (int rb = e0; rb < e1; rb += 32) {
            for (int rr = 0; rr < 32; ++rr) {          // coalesced staging
                const int r = rb + rr;
                tile[rr * TSTR + t] = (r < e1) ? E[(size_t)r * 256 + t] : 0.f;
            }
            __syncthreads();
            const int r = rb + nl;
            if (r < e1) {
                int sid = seg[r] - s0;
                if (sid < 0) sid = 0; if (sid >= SPB) sid = SPB - 1;
                const float* kqc = kq + sid * 8 + h;
                const float* tr  = tile + nl * TSTR;
                float acc = 0.f;
                for (int c = 0; c < 256; ++c) acc += tr[c] * kqc[c * KQSTR];
                const int idx = r - e0;
                if (idx < MAXM) sc[idx * 8 + h] = acc;
            }
            __syncthreads();
        }
    }

    // ---- phase 5: per-(sample,head) stable softmax, in place --------------
    if (t < SPB * 8) {
        const int s = t >> 3, h = t & 7;
        int rs = sstart[s] - e0, re = sstart[s + 1] - e0;
        if (re > MAXM) re = MAXM;
        if (re > rs) {
            float mx = -3.4e38f;
            for (int r = rs; r < re; ++r) mx = fmaxf(mx, sc[r * 8 + h]);
            float den = 0.f;
            for (int r = rs; r < re; ++r) den += expf(sc[r * 8 + h] - mx);
            const float inv = 1.f / den;
            for (int r = rs; r < re; ++r) sc[r * 8 + h] = expf(sc[r * 8 + h] - mx) * inv;
        }
    }
    __syncthreads();

    // ---- phase 6: P[(s,h)][c] = sum_n probs[n,h]*E[n,c] (overlays kq) -----
    {
        const int c = t;
        for (int i = 0; i < SPB; ++i) {
            const int rs = sstart[i], re = sstart[i + 1];
            float acc[8] = {0.f, 0.f, 0.f, 0.f, 0.f, 0.f, 0.f, 0.f};
            for (int r = rs; r < re; ++r) {
                const float e = E[(size_t)r * 256 + c];
                int idx = r - e0; if (idx >= MAXM) idx = MAXM - 1;
#pragma unroll
                for (int h = 0; h < 8; ++h) acc[h] += sc[idx * 8 + h] * e;
            }
#pragma unroll
            for (int h = 0; h < 8; ++h) Pm[(i * 8 + h) * TSTR + c] = acc[h];
        }
    }
    __syncthreads();

    // ---- phase 7: per_sample[s][h*32+j] = P[(s,h),:] @ Wv[:,h*32+j] -------
    float* ps = sm + OFF_R1; // overlays pooled (dead)
    {
        const int mc = (m < SPB) ? m : (SPB - 1);  // clamp: pad rows duplicate
        for (int job = w; job < 16; job += 8) {    // 8 heads x 2 n-subtiles
            const int h = job >> 1, nt = job & 1;
            const int n = h * 32 + nt * 16 + m;
            const float* arow = Pm + (mc * 8 + h) * TSTR + klo;
            v8f acc = {0.f, 0.f, 0.f, 0.f, 0.f, 0.f, 0.f, 0.f};
            for (int kb = 0; kb < 256; kb += 4) {
                v2f a = ld2(arow + kb);             // ds_load_b64, unconditional
                v2f b;
                b.x = Wv[(size_t)(kb + klo) * 256 + n];
                b.y = Wv[(size_t)(kb + klo + 1) * 256 + n];
                acc = wmma_f32_k4(a, b, acc);
            }
#pragma unroll
            for (int j = 0; j < 8; ++j) {          // store all 16 rows; pad rows
                const int row = (lane < 16) ? j : (j + 8); // hold finite junk
                ps[row * TSTR + n] = acc[j];
            }
        }
    }
    __syncthreads();

    // ---- phase 8: out = per_sample @ Wo, store global ---------------------
    for (int nt = w; nt < 16; nt += 8) {
        const int n = nt * 16 + m;
        const float* arow = ps + m * TSTR + klo;
        v8f acc = {0.f, 0.f, 0.f, 0.f, 0.f, 0.f, 0.f, 0.f};
        for (int kb = 0; kb < 256; kb += 4) {
            v2f a = ld2(arow + kb);                 // ds_load_b64, unconditional
            v2f b;
            b.x = Wo[(size_t)(kb + klo) * 256 + n];
            b.y = Wo[(size_t)(kb + klo + 1) * 256 + n];
            acc = wmma_f32_k4(a, b, acc);
        }
#pragma unroll
        for (int j = 0; j < 8; ++j) {
            const int row = (lane < 16) ? j : (j + 8);
            const int s   = s0 + row;
            if (row < SPB && s < S) out[(size_t)s * 256 + n] = acc[j];
        }
    }
}

extern "C" void kernel_launch(void* const* d_in, const int* in_sizes, int n_in,
                              void* d_out, int out_size, void* d_ws, size_t ws_size,
                              hipStream_t stream) {
    (void)n_in; (void)d_ws; (void)ws_size;
    const float* E   = (const float*)d_in[0];
    const int*   seg = (const int*)d_in[1];
    // d_in[2] = num_samples (device scalar) — derived host-side from out_size instead
    const float* Wq  = (const float*)d_in[3];
    const float* Wk  = (const float*)d_in[4];
    const float* Wv  = (const float*)d_in[5];
    const float* Wo  = (const float*)d_in[6];
    float* out = (float*)d_out;

    const int N = in_sizes[0] / 256;
    const int S = out_size / 256;
    const int blocks = (S + SPB - 1) / SPB;

    hipLaunchKernelGGL(msa_segment_attn_kernel, dim3(blocks), dim3(256), 0, stream,
                       E, seg, Wq, Wk, Wv, Wo, out, N, S);
}